// SelectiveSSM_20804821582028
// MI455X (gfx1250) — compile-verified
//
#include <hip/hip_runtime.h>
#include <hip/hip_bf16.h>

#define D_MODEL 1024
#define D_STATE 16
#define D_CONV  4
#define D_INNER 2048
#define BATCH   2
#define SEQ     2048
#define ML      (BATCH * SEQ)          // 4096 rows in all GEMMs
#define NDBL    (2 * D_STATE + 1)      // 33

typedef float v2f __attribute__((ext_vector_type(2)));
typedef float v8f __attribute__((ext_vector_type(8)));
typedef int   v4i __attribute__((ext_vector_type(4)));

#if __has_builtin(__builtin_amdgcn_global_load_async_to_lds_b128) && \
    __has_builtin(__builtin_amdgcn_s_wait_asynccnt)
#define HAVE_ASYNC_LDS 1
typedef __attribute__((address_space(1))) v4i* g128_t;   // global, 16B units
typedef __attribute__((address_space(3))) v4i* l128_t;   // LDS, 16B units
#else
#define HAVE_ASYNC_LDS 0
#endif

__device__ __forceinline__ float fast_sigmoid(float x) {
    return 1.0f / (1.0f + __expf(-x));
}

// ---------------------------------------------------------------------------
// WMMA GEMM:  out[m,n] = sum_k A[m*K + k] * W[n*K + k]
// A: M x K row-major.  W: N x K row-major (multiply by W^T).
// One wave computes a (16*MT) x (16*NT) tile with V_WMMA_F32_16X16X4_F32.
//
// N edge handling: B row pointers are CLAMPED (never predicated). Column n of
// D depends only on column n of B, so garbage in clamped columns only reaches
// output columns that are never stored. This keeps EXEC fully set and branch-
// free in the hot loop (WMMA requires EXEC = all ones anyway).
//
// Fragment layouts per CDNA5 ISA 7.12.2 (32-bit, 16x4 A / 4x16 B / 16x16 C):
//   A: lane<16 -> row m0+lane, VGPR0=K+0, VGPR1=K+1 ; lane>=16 -> K+2, K+3
//   B: mirrored over N (lane r holds column n0+r, same K split)
//   C: VGPR i -> row m0 + i + 8*(lane>=16), col n0 + (lane&15)
// ---------------------------------------------------------------------------
template <int MT, int NT>
__global__ void wmma_gemm_kernel(const float* __restrict__ A,
                                 const float* __restrict__ W,
                                 float* __restrict__ out,
                                 int M, int N, int K) {
    const int wave   = (blockIdx.x * blockDim.x + threadIdx.x) >> 5;
    const int lane   = threadIdx.x & 31;
    const int tilesN = (N + 16 * NT - 1) / (16 * NT);
    const int tilesM = M / (16 * MT);
    if (wave >= tilesM * tilesN) return;          // wave-uniform exit

    const int tm   = wave / tilesN;
    const int tn   = wave - tm * tilesN;
    const int m0   = tm * (16 * MT);
    const int n0   = tn * (16 * NT);
    const int half = lane >> 4;                   // 0: K+0/1   1: K+2/3
    const int r    = lane & 15;

    v8f acc[MT][NT] = {};

    const float* arow[MT];
#pragma unroll
    for (int i = 0; i < MT; ++i)
        arow[i] = A + (size_t)(m0 + i * 16 + r) * K + 2 * half;

    const float* wrow[NT];
#pragma unroll
    for (int j = 0; j < NT; ++j) {
        int n = n0 + j * 16 + r;
        if (n >= N) n = N - 1;                    // clamp: column never stored
        wrow[j] = W + (size_t)n * K + 2 * half;
    }

#pragma unroll 2
    for (int k = 0; k < K; k += 4) {
        v2f a[MT], b[NT];
#pragma unroll
        for (int i = 0; i < MT; ++i) a[i] = *(const v2f*)(arow[i] + k);
#pragma unroll
        for (int j = 0; j < NT; ++j) b[j] = *(const v2f*)(wrow[j] + k);
#pragma unroll
        for (int i = 0; i < MT; ++i)
#pragma unroll
            for (int j = 0; j < NT; ++j)
                acc[i][j] = __builtin_amdgcn_wmma_f32_16x16x4_f32(
                    false, a[i], false, b[j], (short)0, acc[i][j], false, false);
    }

#pragma unroll
    for (int i = 0; i < MT; ++i)
#pragma unroll
        for (int j = 0; j < NT; ++j) {
            const int n = n0 + j * 16 + r;
            if (n < N) {
#pragma unroll
                for (int e = 0; e < 8; ++e) {
                    const int m = m0 + i * 16 + e + 8 * half;
                    out[(size_t)m * N + n] = acc[i][j][e];
                }
            }
        }
}

// ---------------------------------------------------------------------------
// Causal depthwise conv (width 4) + bias + SiLU.
// xz layout: (B*L, 2*D_INNER); xp = columns [0, D_INNER).
// ---------------------------------------------------------------------------
__global__ void conv_silu_kernel(const float* __restrict__ xz,
                                 const float* __restrict__ conv_w,
                                 const float* __restrict__ conv_b,
                                 float* __restrict__ xc) {
    const int idx = blockIdx.x * blockDim.x + threadIdx.x;
    if (idx >= BATCH * SEQ * D_INNER) return;
    const int d = idx & (D_INNER - 1);
    const int l = (idx >> 11) & (SEQ - 1);
    const int b = idx >> 22;

    const float* w = conv_w + d * D_CONV;
    float acc = conv_b[d];
#pragma unroll
    for (int j = 0; j < D_CONV; ++j) {
        const int ls = l - (D_CONV - 1) + j;
        if (ls >= 0)
            acc = fmaf(xz[(size_t)(b * SEQ + ls) * (2 * D_INNER) + d], w[j], acc);
    }
    xc[idx] = acc * fast_sigmoid(acc);
}

// ---------------------------------------------------------------------------
// Selective scan. One thread per (b, d); 16 states in registers.
// Per-timestep shared operands (delta_raw, B_t[16], C_t[16]) staged into LDS
// in chunks of SCAN_T timesteps — via GLOBAL_LOAD_ASYNC_TO_LDS_B128 when the
// toolchain exposes it (ASYNCcnt-tracked), else a synchronous copy.
// All in-loop LDS reads are same-address broadcasts (conflict-free).
// ---------------------------------------------------------------------------
#define SCAN_T 64
__global__ void scan_kernel(const float* __restrict__ xdbl,   // (B*L, 33)
                            const float* __restrict__ xc,     // (B*L, D_INNER)
                            const float* __restrict__ xz,     // (B*L, 2*D_INNER)
                            const float* __restrict__ A_log,  // (D_INNER, 16)
                            float* __restrict__ yb) {         // (B*L, D_INNER)
    __shared__ float sx[SCAN_T * NDBL];                       // 2112 floats
    const int tid = threadIdx.x;                              // 0..127
    const int b   = blockIdx.x >> 4;
    const int d   = ((blockIdx.x & 15) << 7) + tid;

    float a[D_STATE], h[D_STATE];
#pragma unroll
    for (int s = 0; s < D_STATE; ++s) {
        a[s] = -__expf(A_log[d * D_STATE + s]);
        h[s] = 0.0f;
    }

    for (int t0 = 0; t0 < SEQ; t0 += SCAN_T) {
        const size_t chunk_base = (size_t)(b * SEQ + t0) * NDBL;   // 16B aligned
        __syncthreads();
#if HAVE_ASYNC_LDS
        {
            g128_t g = (g128_t)(xdbl + chunk_base);
            l128_t l = (l128_t)sx;
            // SCAN_T*NDBL = 2112 floats = 528 x b128 transfers
            for (int q = tid; q < (SCAN_T * NDBL) / 4; q += 128)
                __builtin_amdgcn_global_load_async_to_lds_b128(g + q, l + q, 0, 0);
            __builtin_amdgcn_s_wait_asynccnt(0);
        }
#else
        for (int i = tid; i < SCAN_T * NDBL; i += 128)
            sx[i] = xdbl[chunk_base + i];
#endif
        __syncthreads();

#pragma unroll 1
        for (int tt = 0; tt < SCAN_T; ++tt) {
            const int t = t0 + tt;
            const float* row = sx + tt * NDBL;
            const float xr    = row[0];
            const float delta = (xr > 20.0f) ? xr : __logf(1.0f + __expf(xr));
            const float u     = xc[(size_t)(b * SEQ + t) * D_INNER + d];
            const float du    = delta * u;
            float y = 0.0f;
#pragma unroll
            for (int s = 0; s < D_STATE; ++s) {
                h[s] = fmaf(__expf(delta * a[s]), h[s], du * row[1 + s]);
                y    = fmaf(h[s], row[1 + D_STATE + s], y);
            }
            const float z = xz[(size_t)(b * SEQ + t) * (2 * D_INNER) + D_INNER + d];
            yb[(size_t)(b * SEQ + t) * D_INNER + d] = y * (z * fast_sigmoid(z));
        }
    }
}

// ---------------------------------------------------------------------------
extern "C" void kernel_launch(void* const* d_in, const int* in_sizes, int n_in,
                              void* d_out, int out_size, void* d_ws, size_t ws_size,
                              hipStream_t stream) {
    (void)in_sizes; (void)n_in; (void)out_size; (void)ws_size;

    const float* x      = (const float*)d_in[0];
    const float* W_in   = (const float*)d_in[1];
    const float* conv_w = (const float*)d_in[2];
    const float* conv_b = (const float*)d_in[3];
    const float* A_log  = (const float*)d_in[4];
    const float* W_x    = (const float*)d_in[5];
    const float* W_out  = (const float*)d_in[6];
    float* out = (float*)d_out;

    // Workspace partition (floats): xz | xc | xdbl | y   (~135 MB total)
    float* xz   = (float*)d_ws;
    float* xc   = xz   + (size_t)ML * (2 * D_INNER);
    float* xdbl = xc   + (size_t)ML * D_INNER;
    float* yb   = xdbl + (size_t)ML * NDBL;

    auto launch_gemm = [&](const float* A, const float* W, float* O,
                           int M, int N, int K) {
        constexpr int MT = 2, NT = 4;
        const int tilesN = (N + 16 * NT - 1) / (16 * NT);
        const int waves  = (M / (16 * MT)) * tilesN;
        const int blocks = (waves + 7) / 8;                    // 8 waves / block
        wmma_gemm_kernel<MT, NT><<<blocks, 256, 0, stream>>>(A, W, O, M, N, K);
    };

    // 1) xz = x @ W_in^T                (4096 x 4096, K=1024)
    launch_gemm(x, W_in, xz, ML, 2 * D_INNER, D_MODEL);
    // 2) xc = silu(depthwise_conv(xp) + b)
    conv_silu_kernel<<<(BATCH * SEQ * D_INNER + 255) / 256, 256, 0, stream>>>(
        xz, conv_w, conv_b, xc);
    // 3) xdbl = xc @ W_x^T              (4096 x 33, K=2048)
    launch_gemm(xc, W_x, xdbl, ML, NDBL, D_INNER);
    // 4) selective scan + SiLU(z) gate -> yb
    scan_kernel<<<BATCH * (D_INNER / 128), 128, 0, stream>>>(xdbl, xc, xz, A_log, yb);
    // 5) out = yb @ W_out^T             (4096 x 1024, K=2048)
    launch_gemm(yb, W_out, out, ML, D_MODEL, D_INNER);
}